// OrthogonalLinear_56564719289139
// MI455X (gfx1250) — compile-verified
//
#include <hip/hip_runtime.h>
#include <math.h>

// OrthogonalLinear on MI455X:
//   out = x @ W + bias,  W = product of 130816 Givens rotations (512x512).
// Phase 1: build W^T on device (rows of W are independent under column rotations).
// Phase 2: 32768x512x512 f32 GEMM via v_wmma_f32_16x16x4_f32, with CDNA5
//          async global->LDS double buffering (global_load_async_to_lds_b128).

#define N_DIM 512
#define NLAYERS 1021   // 2N-3

typedef float v2f __attribute__((ext_vector_type(2)));
typedef float v8f __attribute__((ext_vector_type(8)));
typedef int   v4i __attribute__((ext_vector_type(4)));
typedef __attribute__((address_space(1))) v4i* as1_v4i_p;  // global (__device__)
typedef __attribute__((address_space(3))) v4i* as3_v4i_p;  // LDS (__shared__)

#if defined(__gfx1250__) && __has_builtin(__builtin_amdgcn_global_load_async_to_lds_b128) && \
    __has_builtin(__builtin_amdgcn_s_wait_asynccnt)
#define USE_ASYNC_LDS 1
#else
#define USE_ASYNC_LDS 0
#endif

// ---- wiring closed forms (n=512, m>=n-1 full triangle) ----
// layer d holds wires a = (d&1), (d&1)+2, ..., min(d, 2n-4-d); theta index is
// sequential over (layer, wire-ascending-a).
__device__ __forceinline__ int layer_cnt(int d) {
  return (d <= 510) ? ((d >> 1) + 1) : (255 - ((d - 511) >> 1));
}
__device__ __forceinline__ int layer_base(int d) {
  if (d <= 510) { int q = d >> 1, r = d & 1; return (q + r) * (q + 1); }
  int e = d - 511, p = e >> 1, s = e & 1;
  return 65536 + 255 * e - (p * (p - 1) + s * p);
}

// ---------------- Phase 0: cos/sin of all thetas ----------------
__global__ __launch_bounds__(256) void trig_kernel(const float* __restrict__ th,
                                                   float* __restrict__ cv,
                                                   float* __restrict__ sv, int total) {
  int i = blockIdx.x * 256 + threadIdx.x;
  if (i < total) {
    float t = th[i];
    cv[i] = cosf(t);
    sv[i] = sinf(t);
  }
}

// ---------------- Phase 1: build W^T ----------------
// Each block owns 16 rows of W in LDS; applies all 1021 layers; writes W^T.
#define ROW_STRIDE 516  // 512+4: bank = (4r + a) % 64, distinct for r=0..15
__global__ __launch_bounds__(256) void build_wt_kernel(const float* __restrict__ cosv,
                                                       const float* __restrict__ sinv,
                                                       float* __restrict__ WT) {
  __shared__ float rows[16 * ROW_STRIDE];
  __shared__ float csC[256], csS[256];
  const int tid = threadIdx.x;
  const int i0 = blockIdx.x * 16;  // first global row this block owns

  // identity rows
  for (int idx = tid; idx < 16 * N_DIM; idx += 256) {
    int r = idx >> 9, j = idx & 511;
    rows[r * ROW_STRIDE + j] = (i0 + r == j) ? 1.0f : 0.0f;
  }
  __syncthreads();

  for (int d = 0; d < NLAYERS; ++d) {
    const int cnt = layer_cnt(d);
    const int base = layer_base(d);
    const int amin = d & 1;
    if (tid < cnt) { csC[tid] = cosv[base + tid]; csS[tid] = sinv[base + tid]; }
    __syncthreads();
    const int items = cnt << 4;  // cnt wires x 16 rows
    for (int t = tid; t < items; t += 256) {
      int w = t >> 4, r = t & 15;
      int a = amin + (w << 1);
      float c = csC[w], s = csS[w];
      float* p = &rows[r * ROW_STRIDE + a];
      float x0 = p[0], x1 = p[1];
      p[0] = c * x0 - s * x1;
      p[1] = s * x0 + c * x1;
    }
    __syncthreads();
  }

  // write transposed: WT[j][i0 + r] = W[i0 + r][j]; 16B-aligned float4 chunks
  for (int idx = tid; idx < (N_DIM * 16) / 4; idx += 256) {
    int j = idx >> 2;
    int r4 = (idx & 3) << 2;
    float4 v;
    v.x = rows[(r4 + 0) * ROW_STRIDE + j];
    v.y = rows[(r4 + 1) * ROW_STRIDE + j];
    v.z = rows[(r4 + 2) * ROW_STRIDE + j];
    v.w = rows[(r4 + 3) * ROW_STRIDE + j];
    *(float4*)&WT[(size_t)j * N_DIM + i0 + r4] = v;
  }
}

// ---------------- Phase 2: GEMM out = X @ W + bias ----------------
// Workgroup 256 thr (8 waves): 128x128 output tile; wave: 64x32 (4x2 C-tiles).
// K staged in LDS, 16 per chunk, async double-buffered. f32 WMMA 16x16x4:
//   A lane: m = (lane&15), VGPRs = {X[m][k + 2*(lane>>4)], +1}
//   B lane: n = (lane&15), VGPRs = {WT[n][k + 2*(lane>>4)], +1}
//   C VGPR j, lane: row = j + 8*(lane>>4), col = lane&15
#define KC 16
#define NCHUNK (N_DIM / KC)
#define LSTR 20  // float4-aligned; 20m mod 64 distinct for m=0..15 (bank-clean)
#define TILE_F (128 * LSTR)

__global__ __launch_bounds__(256) void gemm_kernel(const float* __restrict__ X,
                                                   const float* __restrict__ WT,
                                                   const float* __restrict__ bias,
                                                   float* __restrict__ Out, int batch) {
#if USE_ASYNC_LDS
  __shared__ float Xs[2][TILE_F];
  __shared__ float Ws[2][TILE_F];
#else
  __shared__ float Xs[1][TILE_F];
  __shared__ float Ws[1][TILE_F];
#endif
  const int tid = threadIdx.x;
  const int lane = tid & 31;
  const int wid = tid >> 5;
  const int wm = wid & 1;    // wave row 0..1 (64 rows each)
  const int wn = wid >> 1;   // wave col 0..3 (32 cols each)
  const int gm = blockIdx.x;
  const int gn = blockIdx.y;
  const int hl = lane >> 4;  // lane half
  const int l15 = lane & 15;

  v8f acc[4][2];
#pragma unroll
  for (int mt = 0; mt < 4; ++mt)
#pragma unroll
    for (int nt = 0; nt < 2; ++nt) acc[mt][nt] = (v8f){0, 0, 0, 0, 0, 0, 0, 0};

  const float* Xbase = X + (size_t)(gm * 128) * N_DIM;
  const float* Wbase = WT + (size_t)(gn * 128) * N_DIM;

  // per-thread staging coordinates: 512 float4 per 128x16 tile, 2 per thread
  const int r0 = tid >> 2;               // rows tid/4 and tid/4 + 64
  const int c4 = (tid & 3) << 2;         // float4 column offset 0/4/8/12

#if USE_ASYNC_LDS
  // ---- async double-buffered pipeline ----
#define ISSUE_CHUNK(ci, buf)                                                           \
  {                                                                                    \
    const int kc_ = (ci) * KC;                                                         \
    _Pragma("unroll") for (int p = 0; p < 2; ++p) {                                    \
      const int r = r0 + p * 64;                                                       \
      __builtin_amdgcn_global_load_async_to_lds_b128(                                  \
          (as1_v4i_p)(Xbase + (size_t)r * N_DIM + kc_ + c4),                           \
          (as3_v4i_p)&Xs[buf][r * LSTR + c4], 0, 0);                                   \
      __builtin_amdgcn_global_load_async_to_lds_b128(                                  \
          (as1_v4i_p)(Wbase + (size_t)r * N_DIM + kc_ + c4),                           \
          (as3_v4i_p)&Ws[buf][r * LSTR + c4], 0, 0);                                   \
    }                                                                                  \
  }

  ISSUE_CHUNK(0, 0);
  for (int i = 0; i < NCHUNK; ++i) {
    const int cur = i & 1;
    if (i + 1 < NCHUNK) {
      ISSUE_CHUNK(i + 1, cur ^ 1);
      __builtin_amdgcn_s_wait_asynccnt(4);  // chunk i's 4 copies done (in-order)
    } else {
      __builtin_amdgcn_s_wait_asynccnt(0);
    }
    __syncthreads();  // all waves have buffer `cur` populated
#pragma unroll
    for (int ks = 0; ks < 4; ++ks) {
      const int koff = ks * 4 + hl * 2;
      v2f a[4], b[2];
#pragma unroll
      for (int mt = 0; mt < 4; ++mt)
        a[mt] = *(const v2f*)&Xs[cur][(wm * 64 + mt * 16 + l15) * LSTR + koff];
#pragma unroll
      for (int nt = 0; nt < 2; ++nt)
        b[nt] = *(const v2f*)&Ws[cur][(wn * 32 + nt * 16 + l15) * LSTR + koff];
#pragma unroll
      for (int mt = 0; mt < 4; ++mt)
#pragma unroll
        for (int nt = 0; nt < 2; ++nt)
          acc[mt][nt] = __builtin_amdgcn_wmma_f32_16x16x4_f32(
              false, a[mt], false, b[nt], (short)0, acc[mt][nt], false, false);
    }
    __syncthreads();  // done reading `cur` before it is refilled at i+2
  }
#undef ISSUE_CHUNK
#else
  // ---- fallback: single-buffer staged loop (known-good) ----
  for (int kc = 0; kc < N_DIM; kc += KC) {
    __syncthreads();
#pragma unroll
    for (int p = 0; p < 2; ++p) {
      const int r = r0 + p * 64;
      float4 vx = *(const float4*)&Xbase[(size_t)r * N_DIM + kc + c4];
      *(float4*)&Xs[0][r * LSTR + c4] = vx;
      float4 vw = *(const float4*)&Wbase[(size_t)r * N_DIM + kc + c4];
      *(float4*)&Ws[0][r * LSTR + c4] = vw;
    }
    __syncthreads();
#pragma unroll
    for (int ks = 0; ks < 4; ++ks) {
      const int koff = ks * 4 + hl * 2;
      v2f a[4], b[2];
#pragma unroll
      for (int mt = 0; mt < 4; ++mt)
        a[mt] = *(const v2f*)&Xs[0][(wm * 64 + mt * 16 + l15) * LSTR + koff];
#pragma unroll
      for (int nt = 0; nt < 2; ++nt)
        b[nt] = *(const v2f*)&Ws[0][(wn * 32 + nt * 16 + l15) * LSTR + koff];
#pragma unroll
      for (int mt = 0; mt < 4; ++mt)
#pragma unroll
        for (int nt = 0; nt < 2; ++nt)
          acc[mt][nt] = __builtin_amdgcn_wmma_f32_16x16x4_f32(
              false, a[mt], false, b[nt], (short)0, acc[mt][nt], false, false);
    }
  }
#endif

  // epilogue: + bias, store
#pragma unroll
  for (int nt = 0; nt < 2; ++nt) {
    const int ncol = gn * 128 + wn * 32 + nt * 16 + l15;
    const float bv = bias[ncol];
#pragma unroll
    for (int mt = 0; mt < 4; ++mt) {
#pragma unroll
      for (int j = 0; j < 8; ++j) {
        int mrow = gm * 128 + wm * 64 + mt * 16 + j + hl * 8;
        Out[(size_t)mrow * N_DIM + ncol] = acc[mt][nt][j] + bv;
      }
    }
  }
}

extern "C" void kernel_launch(void* const* d_in, const int* in_sizes, int n_in,
                              void* d_out, int out_size, void* d_ws, size_t ws_size,
                              hipStream_t stream) {
  const float* x = (const float*)d_in[0];
  const float* th = (const float*)d_in[1];
  const float* bias = (const float*)d_in[2];
  float* out = (float*)d_out;
  const int batch = in_sizes[0] / N_DIM;  // 32768
  const int total = in_sizes[1];          // 130816

  float* cosv = (float*)d_ws;
  float* sinv = cosv + 131072;           // 512KB offset
  float* WT = cosv + 262144;             // 1MB offset, 512x512 floats

  trig_kernel<<<(total + 255) / 256, 256, 0, stream>>>(th, cosv, sinv, total);
  build_wt_kernel<<<32, 256, 0, stream>>>(cosv, sinv, WT);
  gemm_kernel<<<dim3(batch / 128, N_DIM / 128), dim3(256), 0, stream>>>(x, WT, bias, out, batch);
}